// LocalGrouper_41274635714647
// MI455X (gfx1250) — compile-verified
//
#include <hip/hip_runtime.h>

#define BATCH  16
#define NPTS   4096
#define CDIM   5
#define GROUPS 1024
#define KNN    32
#define CHAN   64
#define FEAT   69    // CHAN + CDIM
#define OUTC   133   // FEAT + CHAN

typedef __attribute__((ext_vector_type(2))) float v2f;
typedef __attribute__((ext_vector_type(8))) float v8f;

// ---------------------------------------------------------------------------
// Kernel 1: farthest point sampling. One workgroup (512 threads = 16 waves)
// per batch. Each thread keeps its 8 points' coords AND running min-distances
// in registers; per-iteration argmax = 5 __shfl_xor butterfly steps per wave
// + a 16-entry LDS combine. Only 2 barriers per sequential iteration.
// Tie-break = lowest index, matching jnp.argmax.
// ---------------------------------------------------------------------------
__global__ __launch_bounds__(512) void fps_kernel(const float* __restrict__ xyz,
                                                  int* __restrict__ fps_idx,
                                                  float* __restrict__ new_xyz) {
  const int b    = blockIdx.x;
  const int tid  = threadIdx.x;
  const int lane = tid & 31;
  const int wid  = tid >> 5;       // 16 waves
  __shared__ float wv[16];
  __shared__ int   wi[16];
  __shared__ int   s_far;

  const float* xb = xyz + (size_t)b * NPTS * CDIM;
  const int base = tid * 8;        // contiguous ownership -> ascending-index ties

  float px[8][CDIM];
  float dist[8];
  #pragma unroll
  for (int j = 0; j < 8; ++j) {
    #pragma unroll
    for (int c = 0; c < CDIM; ++c) px[j][c] = xb[(base + j) * CDIM + c];
    dist[j] = 1e10f;
  }
  if (tid == 0) s_far = 0;
  __syncthreads();

  for (int it = 0; it < GROUPS; ++it) {
    const int far = s_far;
    // centroid: uniform-address loads, L0/L2-hot (batch slice is 80 KB)
    const float c0 = xb[far * CDIM + 0];
    const float c1 = xb[far * CDIM + 1];
    const float c2 = xb[far * CDIM + 2];
    const float c3 = xb[far * CDIM + 3];
    const float c4 = xb[far * CDIM + 4];
    if (tid == 0) {
      fps_idx[b * GROUPS + it] = far;
      float* o = new_xyz + (size_t)(b * GROUPS + it) * CDIM;
      o[0] = c0; o[1] = c1; o[2] = c2; o[3] = c3; o[4] = c4;
    }

    float bestv = -1.0f;
    int   besti = 0;
    #pragma unroll
    for (int j = 0; j < 8; ++j) {
      float d0 = px[j][0] - c0, d1 = px[j][1] - c1, d2 = px[j][2] - c2,
            d3 = px[j][3] - c3, d4 = px[j][4] - c4;
      float d  = d0 * d0 + d1 * d1 + d2 * d2 + d3 * d3 + d4 * d4;
      float nd = fminf(dist[j], d);
      dist[j] = nd;
      if (nd > bestv) { bestv = nd; besti = base + j; }  // ascending j keeps lowest idx
    }

    // intra-wave butterfly argmax (wave32)
    #pragma unroll
    for (int m = 16; m >= 1; m >>= 1) {
      float ov = __shfl_xor(bestv, m, 32);
      int   oi = __shfl_xor(besti, m, 32);
      if (ov > bestv || (ov == bestv && oi < besti)) { bestv = ov; besti = oi; }
    }
    if (lane == 0) { wv[wid] = bestv; wi[wid] = besti; }
    __syncthreads();
    if (tid == 0) {
      float bv = wv[0]; int bi = wi[0];
      #pragma unroll
      for (int w = 1; w < 16; ++w) {
        if (wv[w] > bv || (wv[w] == bv && wi[w] < bi)) { bv = wv[w]; bi = wi[w]; }
      }
      s_far = bi;
    }
    __syncthreads();
  }
}

// ---------------------------------------------------------------------------
// Kernel 2: KNN via WMMA. One wave handles a 16-query strip.
// score[s,n] = |p_n|^2 - 2 q_s . p_n   (|q|^2 dropped: ordering-invariant).
// Cross term via two V_WMMA_F32_16X16X4_F32 (K = 0..3 and 4..7, coords 5..7
// zero-padded, -2 folded into A). Accumulator pre-loaded with |p_n|^2.
// A layout (16x4 f32): lanes 0-15 hold K=0(.x),1(.y); lanes 16-31 K=2,3.
// C layout: VGPR j -> row M=j (lanes 0-15) / M=j+8 (lanes 16-31), col N=lane&15.
//
// Selection: two lanes per query row (lane r: cols 0-7, lane r+16: cols 8-15).
// Each lane streams its 2048 candidates through an UNSORTED register-resident
// top-32 buffer with a running (curMax, maxSlot): rejection = 1 compare;
// replacement = unrolled cndmask write + unrolled max-rescan (branchless,
// no exec-mask chains, no LDS in the hot path). Final: dump both 32-buffers
// per query to LDS and run 32 min-extract rounds (lowest-index tie) to get
// lax.top_k's ascending order.
// ---------------------------------------------------------------------------
__global__ __launch_bounds__(32) void knn_kernel(const float* __restrict__ xyz,
                                                 const float* __restrict__ new_xyz,
                                                 int* __restrict__ knn_idx) {
  const int b    = blockIdx.y;
  const int s0   = blockIdx.x * 16;
  const int lane = threadIdx.x;
  const int half = lane >> 4;   // 0: K rows 0-1 / C rows 0-7 ; 1: K rows 2-3 / C rows 8-15
  const int l16  = lane & 15;

  __shared__ float tile[16 * 16];
  __shared__ float md[16 * 2 * KNN];
  __shared__ int   mi[16 * 2 * KNN];

  // A operand: query row m = s0 + l16, scaled by -2.
  const float* qrow = new_xyz + ((size_t)b * GROUPS + s0 + l16) * CDIM;
  const float q0 = qrow[0], q1 = qrow[1], q2 = qrow[2], q3 = qrow[3], q4 = qrow[4];
  v2f a1, a2;
  a1.x = -2.0f * (half ? q2 : q0);
  a1.y = -2.0f * (half ? q3 : q1);
  a2.x = half ? 0.0f : (-2.0f * q4);
  a2.y = 0.0f;

  // register-resident unsorted top-32 buffer
  float bd[KNN];
  int   bx[KNN];
  #pragma unroll
  for (int j = 0; j < KNN; ++j) { bd[j] = 3.0e38f; bx[j] = 0; }
  float curMax  = 3.0e38f;
  int   maxSlot = 0;

  const float* xb = xyz + (size_t)b * NPTS * CDIM;

  for (int t = 0; t < NPTS / 16; ++t) {
    const int n = t * 16 + l16;
    const float* prow = xb + (size_t)n * CDIM;
    __builtin_prefetch(prow + 16 * CDIM, 0, 1);  // next tile (speculative, OOB-safe)
    const float p0 = prow[0], p1 = prow[1], p2 = prow[2], p3 = prow[3], p4 = prow[4];
    const float pn = p0 * p0 + p1 * p1 + p2 * p2 + p3 * p3 + p4 * p4;

    v2f b1, b2;
    b1.x = half ? p2 : p0;
    b1.y = half ? p3 : p1;
    b2.x = half ? 0.0f : p4;
    b2.y = 0.0f;

    v8f c;
    #pragma unroll
    for (int j = 0; j < 8; ++j) c[j] = pn;  // accumulate onto |p|^2 (per column)

    c = __builtin_amdgcn_wmma_f32_16x16x4_f32(false, a1, false, b1, (short)0, c, false, false);
    c = __builtin_amdgcn_wmma_f32_16x16x4_f32(false, a2, false, b2, (short)0, c, false, false);

    #pragma unroll
    for (int j = 0; j < 8; ++j) tile[(half * 8 + j) * 16 + l16] = c[j];
    __syncthreads();

    // lane (l16, half) reads its 8 contiguous columns of row l16 (32B span)
    float cv[8];
    #pragma unroll
    for (int cc = 0; cc < 8; ++cc) cv[cc] = tile[l16 * 16 + half * 8 + cc];
    __syncthreads();  // tile reusable next iteration; selection below is reg-only

    #pragma unroll
    for (int cc = 0; cc < 8; ++cc) {
      float v = cv[cc];
      if (v < curMax) {
        const int nidx = t * 16 + half * 8 + cc;
        #pragma unroll
        for (int j = 0; j < KNN; ++j) {
          bool sel = (j == maxSlot);
          bd[j] = sel ? v    : bd[j];
          bx[j] = sel ? nidx : bx[j];
        }
        float m = bd[0]; int ms = 0;
        #pragma unroll
        for (int j = 1; j < KNN; ++j) {
          if (bd[j] > m) { m = bd[j]; ms = j; }
        }
        curMax = m; maxSlot = ms;
      }
    }
  }

  // dump both half-buffers (disjoint candidate sets) and merge per query
  #pragma unroll
  for (int j = 0; j < KNN; ++j) {
    md[(l16 * 2 + half) * KNN + j] = bd[j];
    mi[(l16 * 2 + half) * KNN + j] = bx[j];
  }
  __syncthreads();

  if (half == 0) {
    float* dd = &md[l16 * 2 * KNN];  // 64 entries for query row l16
    int*   ii = &mi[l16 * 2 * KNN];
    int*   outp = knn_idx + ((size_t)(b * GROUPS + s0 + l16)) * KNN;
    for (int k = 0; k < KNN; ++k) {
      float bv = dd[0]; int bidx = ii[0]; int bpos = 0;
      for (int e = 1; e < 2 * KNN; ++e) {
        float v = dd[e]; int ix = ii[e];
        if (v < bv || (v == bv && ix < bidx)) { bv = v; bidx = ix; bpos = e; }
      }
      outp[k] = bidx;
      dd[bpos] = 3.0e38f;
    }
  }
}

// ---------------------------------------------------------------------------
// Kernel 3: per-(b,s,c) mean over K, plus per-group sum of centered^2
// (centered values sum to zero, so std^2 = sum(centered^2)/(n-1)).
// One WG per (b,s); thread t handles channel c = t (< 69).
// ---------------------------------------------------------------------------
__global__ __launch_bounds__(128) void stats_kernel(const float* __restrict__ xyz,
                                                    const float* __restrict__ points,
                                                    const int* __restrict__ knn_idx,
                                                    float* __restrict__ mean,
                                                    float* __restrict__ partial) {
  const int wg = blockIdx.x;  // b*GROUPS + s
  const int b  = wg / GROUPS;
  const int t  = threadIdx.x;
  __shared__ float red[128];

  float sq = 0.0f;
  if (t < FEAT) {
    const int* idx = knn_idx + (size_t)wg * KNN;
    float v[KNN];
    float sum = 0.0f;
    #pragma unroll
    for (int k = 0; k < KNN; ++k) {
      int n = idx[k];
      float val = (t < CHAN) ? points[((size_t)b * NPTS + n) * CHAN + t]
                             : xyz[((size_t)b * NPTS + n) * CDIM + (t - CHAN)];
      v[k] = val;
      sum += val;
    }
    float m = sum * (1.0f / KNN);
    mean[(size_t)wg * FEAT + t] = m;
    #pragma unroll
    for (int k = 0; k < KNN; ++k) { float d = v[k] - m; sq += d * d; }
  }
  red[t] = sq;
  __syncthreads();
  for (int off = 64; off > 0; off >>= 1) {
    if (t < off) red[t] += red[t + off];
    __syncthreads();
  }
  if (t == 0) partial[wg] = red[0];
}

// ---------------------------------------------------------------------------
// Kernel 4: per-batch std (ddof=1) from group partials.
// ---------------------------------------------------------------------------
__global__ __launch_bounds__(256) void std_kernel(const float* __restrict__ partial,
                                                  float* __restrict__ stdv) {
  const int b = blockIdx.x;
  const int t = threadIdx.x;
  __shared__ float red[256];
  float s = 0.0f;
  for (int i = t; i < GROUPS; i += 256) s += partial[b * GROUPS + i];
  red[t] = s;
  __syncthreads();
  for (int off = 128; off > 0; off >>= 1) {
    if (t < off) red[t] += red[t + off];
    __syncthreads();
  }
  if (t == 0) stdv[b] = sqrtf(red[0] / (float)((size_t)GROUPS * KNN * FEAT - 1));
}

// ---------------------------------------------------------------------------
// Kernel 5: normalize + affine + anchor concat -> new_points (280 MB write).
// ---------------------------------------------------------------------------
__global__ __launch_bounds__(256) void finalize_kernel(const float* __restrict__ xyz,
                                                       const float* __restrict__ points,
                                                       const float* __restrict__ alpha,
                                                       const float* __restrict__ beta,
                                                       const int* __restrict__ fps_idx,
                                                       const int* __restrict__ knn_idx,
                                                       const float* __restrict__ mean,
                                                       const float* __restrict__ stdv,
                                                       float* __restrict__ out_np) {
  size_t gid = (size_t)blockIdx.x * blockDim.x + threadIdx.x;
  const size_t total = (size_t)BATCH * GROUPS * KNN * OUTC;
  if (gid >= total) return;
  int c = (int)(gid % OUTC);
  size_t tmp = gid / OUTC;
  int k = (int)(tmp % KNN); tmp /= KNN;
  int s = (int)(tmp % GROUPS);
  int b = (int)(tmp / GROUPS);

  float r;
  if (c < FEAT) {
    int n = knn_idx[((size_t)(b * GROUPS + s)) * KNN + k];
    float v = (c < CHAN) ? points[((size_t)b * NPTS + n) * CHAN + c]
                         : xyz[((size_t)b * NPTS + n) * CDIM + (c - CHAN)];
    float m   = mean[((size_t)(b * GROUPS + s)) * FEAT + c];
    float inv = 1.0f / (stdv[b] + 1e-5f);
    r = alpha[c] * (v - m) * inv + beta[c];
  } else {
    int g = fps_idx[b * GROUPS + s];
    r = points[((size_t)b * NPTS + g) * CHAN + (c - FEAT)];
  }
  out_np[gid] = r;
}

// ---------------------------------------------------------------------------
extern "C" void kernel_launch(void* const* d_in, const int* in_sizes, int n_in,
                              void* d_out, int out_size, void* d_ws, size_t ws_size,
                              hipStream_t stream) {
  const float* xyz    = (const float*)d_in[0];
  const float* points = (const float*)d_in[1];
  const float* alpha  = (const float*)d_in[4];
  const float* beta   = (const float*)d_in[5];

  float* out        = (float*)d_out;
  float* new_xyz    = out;                                    // B*GROUPS*CDIM
  float* new_points = out + (size_t)BATCH * GROUPS * CDIM;    // B*GROUPS*KNN*OUTC

  char* ws = (char*)d_ws;
  int*   fps_idx = (int*)ws;   ws += sizeof(int)   * (size_t)BATCH * GROUPS;
  int*   knn_idx = (int*)ws;   ws += sizeof(int)   * (size_t)BATCH * GROUPS * KNN;
  float* mean    = (float*)ws; ws += sizeof(float) * (size_t)BATCH * GROUPS * FEAT;
  float* partial = (float*)ws; ws += sizeof(float) * (size_t)BATCH * GROUPS;
  float* stdv    = (float*)ws;

  fps_kernel<<<BATCH, 512, 0, stream>>>(xyz, fps_idx, new_xyz);
  knn_kernel<<<dim3(GROUPS / 16, BATCH), 32, 0, stream>>>(xyz, new_xyz, knn_idx);
  stats_kernel<<<BATCH * GROUPS, 128, 0, stream>>>(xyz, points, knn_idx, mean, partial);
  std_kernel<<<BATCH, 256, 0, stream>>>(partial, stdv);
  const size_t total = (size_t)BATCH * GROUPS * KNN * OUTC;
  finalize_kernel<<<(unsigned)((total + 255) / 256), 256, 0, stream>>>(
      xyz, points, alpha, beta, fps_idx, knn_idx, mean, stdv, new_points);
}